// GraphSAGE_84524956385806
// MI455X (gfx1250) — compile-verified
//
#include <hip/hip_runtime.h>

#define NODES 100000
#define FEAT 128            // hidden/input feature width (all scatter inputs are 128-wide)
#define EPSV 1e-5f

typedef __attribute__((ext_vector_type(2))) float v2f;
typedef __attribute__((ext_vector_type(8))) float v8f;

__device__ __forceinline__ float atomic_fadd(float* p, float v) {
  return __hip_atomic_fetch_add(p, v, __ATOMIC_RELAXED, __HIP_MEMORY_SCOPE_AGENT);
}

// ---------------- degree: deg[dst] += 1 over all edges ----------------
__global__ void degree_kernel(const int* __restrict__ dst, float* __restrict__ deg, int E) {
  int e = blockIdx.x * blockDim.x + threadIdx.x;
  if (e < E) atomic_fadd(&deg[dst[e]], 1.0f);
}

// ---------------- scatter: agg[dst] += X[src], one wave per edge ----------------
__global__ void scatter_kernel(const float* __restrict__ X, const int* __restrict__ src,
                               const int* __restrict__ dst, float* __restrict__ agg, int E) {
  int wave = (int)((blockIdx.x * (size_t)blockDim.x + threadIdx.x) >> 5);
  int lane = threadIdx.x & 31;
  if (wave >= E) return;
  int s = src[wave];
  int d = dst[wave];
  const float4 v = *(const float4*)&X[(size_t)s * FEAT + lane * 4];
  float* a = &agg[(size_t)d * FEAT + lane * 4];
  atomic_fadd(a + 0, v.x);
  atomic_fadd(a + 1, v.y);
  atomic_fadd(a + 2, v.z);
  atomic_fadd(a + 3, v.w);
}

// ---------------- fused SAGE GEMM:  H = mean(agg)@Wl + X@Wr + b  (+ col stats) ----------------
// One wave computes a 16xC output strip (C/16 tiles of 16x16) via V_WMMA_F32_16X16X4_F32, K=128.
template <int C>
__global__ __launch_bounds__(256) void sage_gemm_kernel(
    const float* __restrict__ agg, const float* __restrict__ deg,
    const float* __restrict__ X,
    const float* __restrict__ Wl, const float* __restrict__ Wr,
    const float* __restrict__ bias,
    float* __restrict__ H, float* __restrict__ colsum, float* __restrict__ colsq) {
  const int K = FEAT;
  const int wave = threadIdx.x >> 5;
  const int lane = threadIdx.x & 31;
  const int mtile = blockIdx.x * 8 + wave;          // wave-uniform: EXEC stays all-ones
  if (mtile >= NODES / 16) return;
  const int m0   = mtile * 16;
  const int mrow = m0 + (lane & 15);                // A-fragment row for this lane
  const int ks   = (lane >> 4) * 2;                 // A: lanes 0-15 -> K{0,1}; 16-31 -> K{2,3}
  const int bs   = (lane >> 4);                     // B: VGPR0 -> K+bs, VGPR1 -> K+2+bs
  const float rdeg = 1.0f / fmaxf(deg[mrow], 1.0f);
  const float* __restrict__ xrow = X   + (size_t)mrow * K;
  const float* __restrict__ arow = agg + (size_t)mrow * K;

  for (int ct = 0; ct < C / 16; ++ct) {
    const int col = ct * 16 + (lane & 15);
    const float bv = bias[col];
    v8f acc = {bv, bv, bv, bv, bv, bv, bv, bv};
#pragma unroll
    for (int k = 0; k < K; k += 4) {
      // X @ Wr
      v2f a;
      { const float2 t = *(const float2*)&xrow[k + ks]; a[0] = t.x; a[1] = t.y; }
      v2f b;
      b[0] = Wr[(k + bs) * C + col];
      b[1] = Wr[(k + 2 + bs) * C + col];
      acc = __builtin_amdgcn_wmma_f32_16x16x4_f32(false, a, false, b, (short)0, acc,
                                                  false, false);
      // mean @ Wl  (mean materialized on the fly: agg * 1/max(deg,1))
      { const float2 t = *(const float2*)&arow[k + ks]; a[0] = t.x * rdeg; a[1] = t.y * rdeg; }
      b[0] = Wl[(k + bs) * C + col];
      b[1] = Wl[(k + 2 + bs) * C + col];
      acc = __builtin_amdgcn_wmma_f32_16x16x4_f32(false, a, false, b, (short)0, acc,
                                                  false, false);
    }
    // D layout: VGPR r -> row m0 + r + (lane>>4)*8, column 'col'
    float s = 0.0f, sq = 0.0f;
#pragma unroll
    for (int r = 0; r < 8; ++r) {
      const int row = m0 + r + (lane >> 4) * 8;
      H[(size_t)row * C + col] = acc[r];
      s  += acc[r];
      sq += acc[r] * acc[r];
    }
    if (colsum != nullptr) {          // fused BN statistics (skipped for final layer)
      atomic_fadd(&colsum[col], s);
      atomic_fadd(&colsq[col], sq);
    }
  }
}

// ---------------- fused BatchNorm + ReLU + dropout (in-place) ----------------
__device__ __forceinline__ unsigned hash_u32(unsigned x) {
  x ^= x >> 16; x *= 0x7feb352du;
  x ^= x >> 15; x *= 0x846ca68bu;
  x ^= x >> 16;
  return x;
}

__global__ void bn_relu_drop_kernel(float* __restrict__ H,
                                    const float* __restrict__ colsum,
                                    const float* __restrict__ colsq,
                                    const float* __restrict__ gamma,
                                    const float* __restrict__ beta,
                                    unsigned seed) {
  const size_t i = (size_t)blockIdx.x * blockDim.x + threadIdx.x;
  const size_t total = (size_t)NODES * FEAT;
  if (i >= total) return;
  const int col = (int)(i & (FEAT - 1));
  const float invN = 1.0f / (float)NODES;
  const float mu  = colsum[col] * invN;
  const float var = colsq[col] * invN - mu * mu;
  const float x = H[i];
  float y = gamma[col] * (x - mu) * rsqrtf(var + EPSV) + beta[col];
  y = fmaxf(y, 0.0f);
  const unsigned h = hash_u32((unsigned)i ^ seed);
  y = (h & 0x80000000u) ? (y * 2.0f) : 0.0f;   // p = 0.5, scale 1/(1-p)
  H[i] = y;
}

extern "C" void kernel_launch(void* const* d_in, const int* in_sizes, int n_in,
                              void* d_out, int out_size, void* d_ws, size_t ws_size,
                              hipStream_t stream) {
  const float* x   = (const float*)d_in[0];
  const int*   ei  = (const int*)d_in[1];
  const float* W1l = (const float*)d_in[2];
  const float* W1r = (const float*)d_in[3];
  const float* b1  = (const float*)d_in[4];
  const float* g1  = (const float*)d_in[5];
  const float* be1 = (const float*)d_in[6];
  const float* W2l = (const float*)d_in[7];
  const float* W2r = (const float*)d_in[8];
  const float* b2  = (const float*)d_in[9];
  const float* g2  = (const float*)d_in[10];
  const float* be2 = (const float*)d_in[11];
  const float* W3l = (const float*)d_in[12];
  const float* W3r = (const float*)d_in[13];
  const float* b3  = (const float*)d_in[14];

  const int E = in_sizes[1] / 2;
  const int* src = ei;
  const int* dst = ei + E;

  // workspace layout (floats)
  float* ws   = (float*)d_ws;
  float* deg  = ws;                              // N
  float* agg  = deg  + NODES;                    // N*128
  float* bufA = agg  + (size_t)NODES * FEAT;     // N*128
  float* bufB = bufA + (size_t)NODES * FEAT;     // N*128
  float* csum = bufB + (size_t)NODES * FEAT;     // 128
  float* csq  = csum + FEAT;                     // 128

  const int eb   = (E + 255) / 256;              // degree blocks
  const int sb   = (E + 7) / 8;                  // scatter blocks (8 waves/block, wave/edge)
  const int gb   = (NODES / 16 + 7) / 8;         // gemm blocks (8 tiles/block)
  const int bnb  = (int)(((size_t)NODES * FEAT + 255) / 256);

  // degree (shared by all three layers)
  hipMemsetAsync(deg, 0, NODES * sizeof(float), stream);
  degree_kernel<<<eb, 256, 0, stream>>>(dst, deg, E);

  // ---- layer 1 ----
  hipMemsetAsync(agg, 0, (size_t)NODES * FEAT * sizeof(float), stream);
  scatter_kernel<<<sb, 256, 0, stream>>>(x, src, dst, agg, E);
  hipMemsetAsync(csum, 0, 2 * FEAT * sizeof(float), stream);
  sage_gemm_kernel<128><<<gb, 256, 0, stream>>>(agg, deg, x, W1l, W1r, b1, bufA, csum, csq);
  bn_relu_drop_kernel<<<bnb, 256, 0, stream>>>(bufA, csum, csq, g1, be1, 0x9E3779B9u);

  // ---- layer 2 ----
  hipMemsetAsync(agg, 0, (size_t)NODES * FEAT * sizeof(float), stream);
  scatter_kernel<<<sb, 256, 0, stream>>>(bufA, src, dst, agg, E);
  hipMemsetAsync(csum, 0, 2 * FEAT * sizeof(float), stream);
  sage_gemm_kernel<128><<<gb, 256, 0, stream>>>(agg, deg, bufA, W2l, W2r, b2, bufB, csum, csq);
  bn_relu_drop_kernel<<<bnb, 256, 0, stream>>>(bufB, csum, csq, g2, be2, 0x85EBCA6Bu);

  // ---- layer 3 (no BN/ReLU/dropout, C = 16, straight to d_out) ----
  hipMemsetAsync(agg, 0, (size_t)NODES * FEAT * sizeof(float), stream);
  scatter_kernel<<<sb, 256, 0, stream>>>(bufB, src, dst, agg, E);
  sage_gemm_kernel<16><<<gb, 256, 0, stream>>>(agg, deg, bufB, W3l, W3r, b3,
                                               (float*)d_out, nullptr, nullptr);
}